// SES_10007273799976
// MI455X (gfx1250) — compile-verified
//
#include <hip/hip_runtime.h>
#include <hip/hip_bf16.h>

// ---------------------------------------------------------------------------
// Simple exponential smoothing as a constant-coefficient first-order scan:
//   level_t = a*level_{t-1} + b_t ,  a = 1-alpha, b_t = alpha*y_t
// Trick 1: since a = 1-alpha, a fictitious incoming carry F_{-1} = y_0 with
//   the uniform rule b_t = alpha*y_t pins level_0 = a*y_0 + alpha*y_0 = y_0.
// Trick 2: WMMA is linear, so M x (alpha*Y) = (alpha*M) x Y -> fold alpha into
//   the tile-invariant A-fragments and stage RAW y bytes in LDS. This lets
//   the input flow global -> LDS via GLOBAL_LOAD_ASYNC_TO_LDS_B128 (ASYNCcnt),
//   double-buffered so the DMA of tile t+1 overlaps the WMMA of tile t.
//
// 3-kernel scheme (64MB input fits in 192MB L2, so the extra pass is ~free):
//   K1: per-4096-elem segment aggregate  E_s = sum a^{SEG-1-i} alpha*y_i
//   K2: serial scan of aggregates, carry_0 = y_0 -> carry into each segment
//   K3: per-wave WMMA tile scan: 16x16 tiles as D = (aM) x Y with
//       V_WMMA_F32_16X16X4_F32 x4 (K accumulation), v_readlane carry chain,
//       async global->LDS input, b128 LDS + global stores, stride-20 padding.
// ---------------------------------------------------------------------------

typedef __attribute__((ext_vector_type(2))) float v2f;
typedef __attribute__((ext_vector_type(8))) float v8f;

#define SEG   4096   // elements per wave-owned segment
#define TILE  256    // 16x16 WMMA tile
#define NTL   (SEG / TILE)
#define WPB   8      // waves per 256-thread block
#define LDSW  320    // 20*16 padded tile stage (16B-aligned float4,
                     // 20n mod 64 distinct for n=0..15 -> conflict-free)

__device__ __forceinline__ float powa32(float a, int d) {
    // a^d for 0 <= d <= 31, uniform-bound loop
    float p = 1.0f;
    #pragma unroll
    for (int i = 0; i < 31; ++i) p = (i < d) ? p * a : p;
    return p;
}

__device__ __forceinline__ void async_ld128(const float* gp, const float* lp) {
    unsigned la = (unsigned)(uintptr_t)lp;          // low 32 bits = LDS offset
    asm volatile("global_load_async_to_lds_b128 %0, %1, off"
                 :: "v"(la), "v"(gp) : "memory");
}

// ---------------- Kernel 1: segment aggregates ----------------
__global__ void ses_partials(const float* __restrict__ y,
                             const float* __restrict__ alphap,
                             float* __restrict__ E, int T, int NS) {
    const int lane = threadIdx.x & 31;
    const int wave = (blockIdx.x * blockDim.x + threadIdx.x) >> 5;
    if (wave >= NS) return;

    const float alpha = alphap[0];
    const float a  = 1.0f - alpha;
    const float a2 = a * a, a3 = a2 * a, a4 = a2 * a2;
    float a128 = a4;
    #pragma unroll
    for (int i = 0; i < 5; ++i) a128 = a128 * a128;         // a^128

    const long long base = (long long)wave * SEG;
    // lane handles float4 chunks at i = c*128 + lane*4 + q, weight a^(SEG-1-i).
    float w3 = alpha * powa32(a4, 31 - lane);               // chunk c = SEG/128-1
    float partial = 0.0f;
    for (int c = SEG / 128 - 1; c >= 0; --c) {
        long long idx = base + (long long)c * 128 + lane * 4;
        float4 v;
        if (idx + 3 < T) {
            v = *(const float4*)(y + idx);
        } else {
            v.x = (idx     < T) ? y[idx]     : 0.0f;
            v.y = (idx + 1 < T) ? y[idx + 1] : 0.0f;
            v.z = (idx + 2 < T) ? y[idx + 2] : 0.0f;
            v.w = (idx + 3 < T) ? y[idx + 3] : 0.0f;
        }
        float inner = fmaf(a, v.z, v.w);
        inner = fmaf(a2, v.y, inner);
        inner = fmaf(a3, v.x, inner);
        partial = fmaf(w3, inner, partial);
        w3 *= a128;
    }
    #pragma unroll
    for (int off = 16; off > 0; off >>= 1)
        partial += __shfl_xor(partial, off, 32);
    if (lane == 0) E[wave] = partial;
}

// ---------------- Kernel 2: serial scan of segment aggregates ----------------
__global__ void ses_blockscan(const float* __restrict__ E,
                              const float* __restrict__ y,
                              const float* __restrict__ alphap,
                              float* __restrict__ C, int NS) {
    if (blockIdx.x == 0 && threadIdx.x == 0) {
        const float a = 1.0f - alphap[0];
        float p = a;
        #pragma unroll
        for (int i = 0; i < 12; ++i) p = p * p;             // a^4096 = a^SEG
        float carry = y[0];                                  // fictitious F_{-1}
        for (int s = 0; s < NS; ++s) {
            C[s] = carry;                                    // carry INTO segment s
            carry = fmaf(p, carry, E[s]);
        }
    }
}

// ---------------- Kernel 3: WMMA apply pass ----------------
__global__ void ses_apply(const float* __restrict__ y,
                          const float* __restrict__ alphap,
                          const float* __restrict__ C,
                          float* __restrict__ out, int T, int outN, int NS) {
    __shared__ __align__(16) float lds[WPB * 2 * LDSW];      // double buffered
    const int lane = threadIdx.x & 31;
    const int wib  = threadIdx.x >> 5;                       // wave in block
    const int wave = (blockIdx.x * blockDim.x + threadIdx.x) >> 5;
    if (wave >= NS) return;                                  // uniform wave exit
    float* st0 = &lds[wib * 2 * LDSW];
    float* st1 = st0 + LDSW;

    const float alpha = alphap[0];
    const float a = 1.0f - alpha;
    float p16 = a;
    #pragma unroll
    for (int i = 0; i < 4; ++i) p16 = p16 * p16;             // a^16

    const int m  = lane & 15;
    const int hi = (lane >= 16) ? 1 : 0;

    // A-fragments: alpha * M, M[i][k] = a^(i-k) for i>=k. Tile-invariant.
    // 32-bit 16x4 A layout: lanes0-15 hold (M=lane, K=0/1 in v0/v1),
    //                       lanes16-31 hold (M=lane-16, K=2/3 in v0/v1).
    v2f afrag[4];
    #pragma unroll
    for (int ko = 0; ko < 4; ++ko) {
        int kA = 4 * ko + 2 * hi;
        afrag[ko][0] = (m >= kA)     ? alpha * powa32(a, m - kA)     : 0.0f;
        afrag[ko][1] = (m >= kA + 1) ? alpha * powa32(a, m - kA - 1) : 0.0f;
    }
    // a^(row+1) for this lane's 8 C/D rows (row = r + 8*hi)
    float pr[8];
    pr[0] = powa32(a, 8 * hi + 1);
    #pragma unroll
    for (int r = 1; r < 8; ++r) pr[r] = pr[r - 1] * a;

    float F = C[wave];                                       // carry into segment
    const long long base = (long long)wave * SEG;
    const int li0 = lane * 4;                                // lane's float4 slot
    const int n0  = li0 >> 4, k0 = li0 & 15;                 // j=0 LDS slot
    const int n1  = (128 + li0) >> 4, k1 = li0 & 15;         // j=1 LDS slot
    const long long idxMax = (long long)T - 4;               // clamp (EXEC full)

    // prime the pipeline: async-load tile 0 into st0
    {
        long long i0 = base + li0;              if (i0 > idxMax) i0 = idxMax;
        long long i1 = base + 128 + li0;        if (i1 > idxMax) i1 = idxMax;
        async_ld128(y + i0, st0 + 20 * n0 + k0);
        async_ld128(y + i1, st1 - LDSW + 20 * n1 + k1);      // same buffer st0
    }

    for (int tl = 0; tl < NTL; ++tl) {
        const long long tbase = base + (long long)tl * TILE;
        float* bs = (tl & 1) ? st1 : st0;                    // current tile buf
        float* nb = (tl & 1) ? st0 : st1;                    // next tile buf

        // issue async loads for tile tl+1 (clamped; harmless past segment end)
        {
            long long i0 = tbase + TILE + li0;       if (i0 > idxMax) i0 = idxMax;
            long long i1 = tbase + TILE + 128 + li0; if (i1 > idxMax) i1 = idxMax;
            async_ld128(y + i0, nb + 20 * n0 + k0);
            async_ld128(y + i1, nb + 20 * n1 + k1);
        }
        // tile tl's 2 loads complete when <=2 remain (in-order completion)
        asm volatile("s_wait_asynccnt 0x2" ::: "memory");

        // D = (aM) x Y : four 16x16x4 fp32 WMMAs with K accumulation
        // B layout (mirrors A K-split): v0: lanes0-15 K=4ko+0, lanes16-31 K=4ko+2
        v8f acc = {};
        #pragma unroll
        for (int ko = 0; ko < 4; ++ko) {
            int kB = 4 * ko + 2 * hi;
            v2f bfrag;
            bfrag[0] = bs[20 * m + kB];
            bfrag[1] = bs[20 * m + kB + 1];
            acc = __builtin_amdgcn_wmma_f32_16x16x4_f32(
                false, afrag[ko], false, bfrag, (short)0, acc, false, false);
        }

        // cross-column carry chain: e_n = D[15][n] in acc[7], lane 16+n.
        // Constant source lanes -> v_readlane_b32; running carry wave-uniform.
        float d7 = acc[7];
        float Fcur = F, G = F;
        #pragma unroll
        for (int n = 0; n < 16; ++n) {
            float e = __int_as_float(
                __builtin_amdgcn_readlane(__float_as_int(d7), 16 + n));
            if (m == n) G = Fcur;
            Fcur = fmaf(p16, Fcur, e);
        }
        F = Fcur;                                            // carry out of tile

        // apply carries, stage in time-order (merges to b128), write coalesced
        #pragma unroll
        for (int r = 0; r < 8; r += 2) {
            float2 pairv;
            pairv.x = fmaf(pr[r],     G, acc[r]);
            pairv.y = fmaf(pr[r + 1], G, acc[r + 1]);
            *(float2*)(bs + 20 * m + 8 * hi + r) = pairv;
        }
        asm volatile("s_wait_dscnt 0" ::: "memory");
        #pragma unroll
        for (int j = 0; j < 2; ++j) {
            long long idx = tbase + j * 128 + li0;
            int li = j * 128 + li0;
            float4 v = *(const float4*)(bs + 20 * (li >> 4) + (li & 15));
            if (idx + 3 < outN) {
                *(float4*)(out + idx) = v;
            } else {
                if (idx     < outN) out[idx]     = v.x;
                if (idx + 1 < outN) out[idx + 1] = v.y;
                if (idx + 2 < outN) out[idx + 2] = v.z;
                if (idx + 3 < outN) out[idx + 3] = v.w;
            }
        }
        asm volatile("s_wait_dscnt 0" ::: "memory");         // drain before reuse
    }
    // 2 dangling async loads into LDS are drained by S_ENDPGM's implicit wait.
}

extern "C" void kernel_launch(void* const* d_in, const int* in_sizes, int n_in,
                              void* d_out, int out_size, void* d_ws, size_t ws_size,
                              hipStream_t stream) {
    const float* y      = (const float*)d_in[0];
    const float* alphap = (const float*)d_in[1];
    float*       out    = (float*)d_out;
    const int T  = in_sizes[0];
    const int NS = (T + SEG - 1) / SEG;

    float* E = (float*)d_ws;        // NS floats
    float* C = E + NS;              // NS floats

    const int threads = 32 * WPB;   // 256
    const int blocks  = (NS + WPB - 1) / WPB;

    ses_partials <<<blocks, threads, 0, stream>>>(y, alphap, E, T, NS);
    ses_blockscan<<<1, 32, 0, stream>>>(E, y, alphap, C, NS);
    ses_apply    <<<blocks, threads, 0, stream>>>(y, alphap, C, out, T, out_size, NS);
}